// get_model_27590869909778
// MI455X (gfx1250) — compile-verified
//
#include <hip/hip_runtime.h>

typedef float v2f __attribute__((ext_vector_type(2)));
typedef float v8f __attribute__((ext_vector_type(8)));

#define TPB 256

// ---------------------------------------------------------------------------
// (B,3,N) -> (B,N,3)
__global__ void k_transpose_xyz(const float* __restrict__ in, float* __restrict__ out,
                                int B, int N) {
  long total = (long)B * N;
  for (long i = blockIdx.x * (long)blockDim.x + threadIdx.x; i < total;
       i += (long)gridDim.x * blockDim.x) {
    long b = i / N, n = i % N;
    out[i * 3 + 0] = in[(b * 3 + 0) * N + n];
    out[i * 3 + 1] = in[(b * 3 + 1) * N + n];
    out[i * 3 + 2] = in[(b * 3 + 2) * N + n];
  }
}

// ---------------------------------------------------------------------------
// Farthest point sampling: one workgroup per batch, dist[] in LDS.
__global__ void k_fps(const float* __restrict__ xyz, int* __restrict__ out,
                      int N, int S) {
  __shared__ float dist[4096];
  __shared__ float rv[TPB];
  __shared__ int   ri[TPB];
  __shared__ int   s_far;
  int b = blockIdx.x;
  const float* p = xyz + (long)b * N * 3;
  for (int n = threadIdx.x; n < N; n += blockDim.x) dist[n] = 1e10f;
  int far = 0;
  __syncthreads();
  for (int it = 0; it < S; ++it) {
    if (threadIdx.x == 0) out[b * S + it] = far;
    float cx = p[far * 3 + 0], cy = p[far * 3 + 1], cz = p[far * 3 + 2];
    float bv = -1.0f; int bi = 0;
    for (int n = threadIdx.x; n < N; n += blockDim.x) {
      float dx = p[n * 3 + 0] - cx, dy = p[n * 3 + 1] - cy, dz = p[n * 3 + 2] - cz;
      float d = dx * dx + dy * dy + dz * dz;
      float dm = fminf(dist[n], d);
      dist[n] = dm;
      if (dm > bv || (dm == bv && n < bi)) { bv = dm; bi = n; }
    }
    rv[threadIdx.x] = bv; ri[threadIdx.x] = bi;
    __syncthreads();
    for (int off = blockDim.x >> 1; off > 0; off >>= 1) {
      if (threadIdx.x < (unsigned)off) {
        float ov = rv[threadIdx.x + off]; int oi = ri[threadIdx.x + off];
        if (ov > rv[threadIdx.x] || (ov == rv[threadIdx.x] && oi < ri[threadIdx.x])) {
          rv[threadIdx.x] = ov; ri[threadIdx.x] = oi;
        }
      }
      __syncthreads();
    }
    if (threadIdx.x == 0) s_far = ri[0];
    __syncthreads();
    far = s_far;
    __syncthreads();
  }
}

// ---------------------------------------------------------------------------
__global__ void k_gather_xyz(const float* __restrict__ xyz, const int* __restrict__ idx,
                             float* __restrict__ out, int N, int S, int total) {
  for (int i = blockIdx.x * blockDim.x + threadIdx.x; i < total;
       i += gridDim.x * blockDim.x) {
    int b = i / S;
    int n = idx[i];
    out[(long)i * 3 + 0] = xyz[((long)b * N + n) * 3 + 0];
    out[(long)i * 3 + 1] = xyz[((long)b * N + n) * 3 + 1];
    out[(long)i * 3 + 2] = xyz[((long)b * N + n) * 3 + 2];
  }
}

// ---------------------------------------------------------------------------
// Ball query: one wave32 per centroid; ballot + popc-prefix emits ascending
// in-radius indices (== reference's sort-then-truncate); pads with first hit.
__global__ void k_ball_query(const float* __restrict__ xyz, const float* __restrict__ nxyz,
                             int* __restrict__ idx, int M, int S, int K, float r2,
                             int total) {
  int w = blockIdx.x * (blockDim.x >> 5) + ((int)threadIdx.x >> 5);
  if (w >= total) return;
  int lane = threadIdx.x & 31;
  int b = w / S;
  const float* p = xyz + (long)b * M * 3;
  float cx = nxyz[(long)w * 3 + 0], cy = nxyz[(long)w * 3 + 1], cz = nxyz[(long)w * 3 + 2];
  int* o = idx + (long)w * K;
  int count = 0, first = -1;
  for (int base = 0; base < M && count < K; base += 32) {
    int n = base + lane;
    bool in = false;
    if (n < M) {
      float dx = p[n * 3 + 0] - cx, dy = p[n * 3 + 1] - cy, dz = p[n * 3 + 2] - cz;
      in = (dx * dx + dy * dy + dz * dz) <= r2;
    }
    unsigned mask = (unsigned)__ballot(in);
    if (first < 0 && mask) first = base + __ffs(mask) - 1;
    if (in) {
      int pos = count + __popc(mask & ((1u << lane) - 1u));
      if (pos < K) o[pos] = n;
    }
    count += __popc(mask);
  }
  if (count > K) count = K;
  if (first < 0) first = 0;
  for (int pos = count + lane; pos < K; pos += 32) o[pos] = first;
}

// ---------------------------------------------------------------------------
// Grouped features with padded row stride Cpad; pad columns zeroed.
__global__ void k_group(const float* __restrict__ xyz, const float* __restrict__ pts,
                        const float* __restrict__ nxyz, const int* __restrict__ idx,
                        float* __restrict__ gp, int M, int S, int K, int Cp, int Cpad,
                        long total) {
  for (long i = blockIdx.x * (long)blockDim.x + threadIdx.x; i < total;
       i += (long)gridDim.x * blockDim.x) {
    long bs = i / K;
    long b  = bs / S;
    int  n  = idx[i];
    const float* src = pts + ((long)b * M + n) * Cp;
    float* d = gp + i * Cpad;
    for (int c = 0; c < Cp; ++c) d[c] = src[c];
    d[Cp + 0] = xyz[((long)b * M + n) * 3 + 0] - nxyz[bs * 3 + 0];
    d[Cp + 1] = xyz[((long)b * M + n) * 3 + 1] - nxyz[bs * 3 + 1];
    d[Cp + 2] = xyz[((long)b * M + n) * 3 + 2] - nxyz[bs * 3 + 2];
    for (int c = Cp + 3; c < Cpad; ++c) d[c] = 0.0f;
  }
}

// ---------------------------------------------------------------------------
// Pack/pad weights: wp[(k>>1)*Cp + c][k&1] = w[k][c] (zeros outside), pad bias.
__global__ void k_pad_w(const float* __restrict__ w, const float* __restrict__ b,
                        float* __restrict__ wp, float* __restrict__ bp,
                        int K, int C, int Kp, int Cp) {
  long total = (long)Kp * Cp;
  for (long i = blockIdx.x * (long)blockDim.x + threadIdx.x; i < total;
       i += (long)gridDim.x * blockDim.x) {
    int k = (int)(i / Cp), c = (int)(i % Cp);
    float v = (k < K && c < C) ? w[(long)k * C + c] : 0.0f;
    wp[(((long)(k >> 1) * Cp + c) << 1) | (k & 1)] = v;
    if (i < Cp) bp[i] = (i < C) ? b[i] : 0.0f;
  }
}

// ---------------------------------------------------------------------------
// WMMA f32 GEMM, fully padded: Y[R,Cp] = X[R,Kp] * W + bias.
// One wave per 16x16 tile; inner loop = 2x global_load_b64 + 1x wmma, no branches.
__global__ void __launch_bounds__(TPB)
k_gemm_bias(const float* __restrict__ X, const float* __restrict__ Wp,
            const float* __restrict__ bp, float* __restrict__ Y,
            int R, int Kp, int Cp) {
  int tilesC = Cp >> 4;
  int tiles  = (R >> 4) * tilesC;
  int wave = blockIdx.x * (blockDim.x >> 5) + ((int)threadIdx.x >> 5);
  if (wave >= tiles) return;
  int tr = wave / tilesC, tc = wave % tilesC;
  int lane = threadIdx.x & 31;
  int hv  = lane >> 4;          // 0/1 selects k pair within quad
  int l16 = lane & 15;
  int row = tr * 16 + l16;      // A row for this lane
  int col = tc * 16 + l16;      // B/C column for this lane
  const float* xr = X  + (long)row * Kp + hv * 2;
  const float* wr = Wp + (((long)hv * Cp + col) << 1);
  v8f acc = {};
  int iters = Kp >> 2;
  for (int t = 0; t < iters; ++t) {
    v2f a  = *(const v2f*)xr;
    v2f bb = *(const v2f*)wr;
    acc = __builtin_amdgcn_wmma_f32_16x16x4_f32(false, a, false, bb,
                                                (short)0, acc, false, false);
    xr += 4;
    wr += (long)Cp * 4;
  }
  float bv = bp[col];
#pragma unroll
  for (int i = 0; i < 8; ++i)
    Y[(long)(tr * 16 + i + hv * 8) * Cp + col] = acc[i] + bv;
}

// ---------------------------------------------------------------------------
// Per-channel BN stats over R rows (stride Cp) -> scale/shift. Block per channel.
__global__ void k_bn_stats(const float* __restrict__ Y, int R, int Cp,
                           const float* __restrict__ g, const float* __restrict__ beta,
                           float* __restrict__ scale, float* __restrict__ shift) {
  __shared__ float sh[TPB], sh2[TPB];
  int c = blockIdx.x;
  float s = 0.0f, s2 = 0.0f;
  for (int r = threadIdx.x; r < R; r += blockDim.x) {
    float v = Y[(long)r * Cp + c];
    s += v; s2 += v * v;
  }
  sh[threadIdx.x] = s; sh2[threadIdx.x] = s2;
  __syncthreads();
  for (int off = blockDim.x >> 1; off > 0; off >>= 1) {
    if (threadIdx.x < (unsigned)off) {
      sh[threadIdx.x]  += sh[threadIdx.x + off];
      sh2[threadIdx.x] += sh2[threadIdx.x + off];
    }
    __syncthreads();
  }
  if (threadIdx.x == 0) {
    float mean = sh[0] / (float)R;
    float var  = sh2[0] / (float)R - mean * mean;
    float sc = g[c] * rsqrtf(var + 1e-5f);
    scale[c] = sc;
    shift[c] = beta[c] - mean * sc;
  }
}

__global__ void k_bn_apply_relu(float* __restrict__ Y, const float* __restrict__ scale,
                                const float* __restrict__ shift, int C, int Cp,
                                long total) {
  for (long i = blockIdx.x * (long)blockDim.x + threadIdx.x; i < total;
       i += (long)gridDim.x * blockDim.x) {
    int c = (int)(i % Cp);
    Y[i] = (c < C) ? fmaxf(0.0f, Y[i] * scale[c] + shift[c]) : 0.0f;
  }
}

// ---------------------------------------------------------------------------
// Max over sample axis: in [BS, K, CpIn] -> out[bs, coff+c] (row stride Ctot).
__global__ void k_maxpool(const float* __restrict__ in, float* __restrict__ out,
                          int K, int C, int CpIn, int Ctot, int coff, long total) {
  for (long i = blockIdx.x * (long)blockDim.x + threadIdx.x; i < total;
       i += (long)gridDim.x * blockDim.x) {
    int  c  = (int)(i % C);
    long bs = i / C;
    float m = -3.0e38f;
    for (int j = 0; j < K; ++j) m = fmaxf(m, in[((long)bs * K + j) * CpIn + c]);
    out[bs * Ctot + coff + c] = m;
  }
}

// ---------------------------------------------------------------------------
// Channel-range copy with optional broadcast over M (srcM==1).
__global__ void k_copy_ch(const float* __restrict__ src, float* __restrict__ dst,
                          int M, int srcM, int Cs, int Ctot, int coff, long total) {
  for (long i = blockIdx.x * (long)blockDim.x + threadIdx.x; i < total;
       i += (long)gridDim.x * blockDim.x) {
    int  c  = (int)(i % Cs);
    long bm = i / Cs;
    long b  = bm / M;
    long m  = bm % M;
    long sm = (srcM == 1) ? 0 : m;
    dst[(b * M + m) * (long)Ctot + coff + c] = src[(b * srcM + sm) * (long)Cs + c];
  }
}

// Zero pad columns [c0, Ctot) over `rows` rows.
__global__ void k_zero_cols(float* __restrict__ dst, int Ctot, int c0, long rows) {
  int w = Ctot - c0;
  long total = rows * (long)w;
  for (long i = blockIdx.x * (long)blockDim.x + threadIdx.x; i < total;
       i += (long)gridDim.x * blockDim.x) {
    long r = i / w;
    int  c = (int)(i % w);
    dst[r * Ctot + c0 + c] = 0.0f;
  }
}

// ---------------------------------------------------------------------------
// 3-NN inverse-distance interpolation; writes C2 channels at dst col offset.
__global__ void k_three_nn_interp(const float* __restrict__ x1, const float* __restrict__ x2,
                                  const float* __restrict__ p2, float* __restrict__ dst,
                                  int M, int S, int C2, int C2stride, int Ctot, int coff,
                                  long total) {
  for (long i = blockIdx.x * (long)blockDim.x + threadIdx.x; i < total;
       i += (long)gridDim.x * blockDim.x) {
    long b = i / M;
    float px = x1[i * 3 + 0], py = x1[i * 3 + 1], pz = x1[i * 3 + 2];
    float d0 = 3.0e38f, d1 = 3.0e38f, d2 = 3.0e38f;
    int   i0 = 0, i1 = 0, i2 = 0;
    const float* q = x2 + b * (long)S * 3;
    for (int s = 0; s < S; ++s) {
      float dx = q[s * 3 + 0] - px, dy = q[s * 3 + 1] - py, dz = q[s * 3 + 2] - pz;
      float d = dx * dx + dy * dy + dz * dz;
      if (d < d0)      { d2 = d1; i2 = i1; d1 = d0; i1 = i0; d0 = d; i0 = s; }
      else if (d < d1) { d2 = d1; i2 = i1; d1 = d; i1 = s; }
      else if (d < d2) { d2 = d;  i2 = s; }
    }
    float w0 = 1.0f / (d0 + 1e-8f), w1 = 1.0f / (d1 + 1e-8f), w2 = 1.0f / (d2 + 1e-8f);
    float ws = w0 + w1 + w2;
    w0 /= ws; w1 /= ws; w2 /= ws;
    const float* r0 = p2 + (b * S + i0) * (long)C2stride;
    const float* r1 = p2 + (b * S + i1) * (long)C2stride;
    const float* r2 = p2 + (b * S + i2) * (long)C2stride;
    float* o = dst + i * (long)Ctot + coff;
    for (int c = 0; c < C2; ++c) o[c] = w0 * r0[c] + w1 * r1[c] + w2 * r2[c];
  }
}

// ===========================================================================
struct Lyr { const float *w, *b, *g, *be; int cin, cout; };
struct PL  { const float *wp, *bp, *g, *be; int Kp, Cp, C; };

extern "C" void kernel_launch(void* const* d_in, const int* in_sizes, int n_in,
                              void* d_out, int out_size, void* d_ws, size_t ws_size,
                              hipStream_t stream) {
  (void)in_sizes; (void)out_size; (void)ws_size;
  if (n_in < 103) return;

  const int B = 8, N = 4096;
  const float* xyz_cf = (const float*)d_in[0];
  auto a16 = [](int x) { return (x + 15) & ~15; };

  // ----- parameter mapping (insertion-order flatten; layer = w,b,g,beta) -----
  int p = 1;
  auto take = [&](int cin, int cout) {
    Lyr L{(const float*)d_in[p], (const float*)d_in[p + 1],
          (const float*)d_in[p + 2], (const float*)d_in[p + 3], cin, cout};
    p += 4;
    return L;
  };
  Lyr sa1r[3][3];
  sa1r[0][0] = take(6, 32);  sa1r[0][1] = take(32, 32);  sa1r[0][2] = take(32, 64);
  sa1r[1][0] = take(6, 64);  sa1r[1][1] = take(64, 64);  sa1r[1][2] = take(64, 128);
  sa1r[2][0] = take(6, 64);  sa1r[2][1] = take(64, 96);  sa1r[2][2] = take(96, 128);
  Lyr sa2r[2][3];
  sa2r[0][0] = take(323, 128); sa2r[0][1] = take(128, 128); sa2r[0][2] = take(128, 256);
  sa2r[1][0] = take(323, 128); sa2r[1][1] = take(128, 196); sa2r[1][2] = take(196, 256);
  Lyr sa3r[3];
  sa3r[0] = take(515, 256); sa3r[1] = take(256, 512); sa3r[2] = take(512, 1024);
  Lyr fp3r[2]; fp3r[0] = take(1536, 256); fp3r[1] = take(256, 256);
  Lyr fp2r[2]; fp2r[0] = take(576, 256);  fp2r[1] = take(256, 128);
  Lyr fp1r[2]; fp1r[0] = take(131, 128);  fp1r[1] = take(128, 128);
  Lyr head1r  = take(128, 128);
  Lyr head2r{(const float*)d_in[p], (const float*)d_in[p + 1], nullptr, nullptr, 128, 128};

  // ----- workspace carve -----
  size_t off = 0;
  auto alloc = [&](size_t bytes) {
    void* q = (char*)d_ws + off;
    off += (bytes + 255) & ~(size_t)255;
    return q;
  };
  auto gs = [](long n) { long b = (n + TPB - 1) / TPB; return dim3((unsigned)(b < 1 ? 1 : b)); };

  // Padded/packed weights (prepared on-stream each call; deterministic).
  auto prep = [&](const Lyr& L) -> PL {
    int Kp = a16(L.cin), Cp = a16(L.cout);
    float* wp = (float*)alloc((size_t)Kp * Cp * 4);
    float* bp = (float*)alloc((size_t)Cp * 4);
    k_pad_w<<<gs((long)Kp * Cp), TPB, 0, stream>>>(L.w, L.b, wp, bp, L.cin, L.cout, Kp, Cp);
    return PL{wp, bp, L.g, L.be, Kp, Cp, L.cout};
  };
  PL sa1m[3][3], sa2m[2][3], sa3m[3], fp3m[2], fp2m[2], fp1m[2], head1, head2;
  for (int i = 0; i < 3; ++i) for (int j = 0; j < 3; ++j) sa1m[i][j] = prep(sa1r[i][j]);
  for (int i = 0; i < 2; ++i) for (int j = 0; j < 3; ++j) sa2m[i][j] = prep(sa2r[i][j]);
  for (int j = 0; j < 3; ++j) sa3m[j] = prep(sa3r[j]);
  for (int j = 0; j < 2; ++j) { fp3m[j] = prep(fp3r[j]); fp2m[j] = prep(fp2r[j]); fp1m[j] = prep(fp1r[j]); }
  head1 = prep(head1r);
  head2 = prep(head2r);

  float* xyz_pt = (float*)alloc((size_t)B * N * 3 * 4);
  int*   fpsIdx = (int*)  alloc((size_t)B * 512 * 4);
  float* l1x    = (float*)alloc((size_t)B * 512 * 3 * 4);
  float* l2x    = (float*)alloc((size_t)B * 128 * 3 * 4);
  int*   bqIdx  = (int*)  alloc((size_t)B * 512 * 128 * 4);
  float* l1p    = (float*)alloc((size_t)B * 512 * 320 * 4);   // 320 % 16 == 0
  float* l2p    = (float*)alloc((size_t)B * 128 * 512 * 4);
  float* l3p    = (float*)alloc((size_t)B * 1024 * 4);
  float* scale  = (float*)alloc(1024 * 4);
  float* shift  = (float*)alloc(1024 * 4);
  float* gp     = (float*)alloc((size_t)B * 128 * 128 * 336 * 4); // max padded input buf
  float* actA   = (float*)alloc((size_t)B * 128 * 128 * 256 * 4); // padded activation, sized for SA2
  float* actB   = (float*)alloc((size_t)B * 128 * 128 * 256 * 4);

  auto run_layer = [&](const float* X, int R, const PL& L, float* Y, bool bn) {
    int tiles = (R >> 4) * (L.Cp >> 4);
    k_gemm_bias<<<dim3((tiles + 7) / 8), TPB, 0, stream>>>(X, L.wp, L.bp, Y, R, L.Kp, L.Cp);
    if (bn) {
      k_bn_stats<<<dim3(L.C), TPB, 0, stream>>>(Y, R, L.Cp, L.g, L.be, scale, shift);
      long tot = (long)R * L.Cp;
      k_bn_apply_relu<<<gs(tot), TPB, 0, stream>>>(Y, scale, shift, L.C, L.Cp, tot);
    }
  };
  auto run_mlp = [&](const float* X0, int R, const PL* Ls, int n) -> const float* {
    const float* X = X0;
    for (int i = 0; i < n; ++i) {
      float* Y = (i & 1) ? actB : actA;
      run_layer(X, R, Ls[i], Y, true);
      X = Y;
    }
    return X;
  };

  // ---------- setup ----------
  k_transpose_xyz<<<gs((long)B * N), TPB, 0, stream>>>(xyz_cf, xyz_pt, B, N);

  // ---------- SA1 (4096 -> 512 centroids, 3 radius branches) ----------
  k_fps<<<dim3(B), TPB, 0, stream>>>(xyz_pt, fpsIdx, N, 512);
  k_gather_xyz<<<gs((long)B * 512), TPB, 0, stream>>>(xyz_pt, fpsIdx, l1x, N, 512, B * 512);
  {
    const float rad[3]  = {0.1f, 0.2f, 0.4f};
    const int   ks[3]   = {32, 64, 128};
    const int   coff[3] = {0, 64, 192};
    for (int br = 0; br < 3; ++br) {
      int K = ks[br], Cpad = a16(6);
      int waves = B * 512;
      k_ball_query<<<dim3((waves + 7) / 8), TPB, 0, stream>>>(
          xyz_pt, l1x, bqIdx, N, 512, K, rad[br] * rad[br], waves);
      long tg = (long)B * 512 * K;
      k_group<<<gs(tg), TPB, 0, stream>>>(xyz_pt, xyz_pt, l1x, bqIdx, gp, N, 512, K, 3,
                                          Cpad, tg);
      int R = B * 512 * K;
      const float* res = run_mlp(gp, R, sa1m[br], 3);
      int C = sa1m[br][2].C, Cp = sa1m[br][2].Cp;
      long tp = (long)B * 512 * C;
      k_maxpool<<<gs(tp), TPB, 0, stream>>>(res, l1p, K, C, Cp, 320, coff[br], tp);
    }
  }

  // ---------- SA2 (512 -> 128 centroids, 2 radius branches) ----------
  k_fps<<<dim3(B), TPB, 0, stream>>>(l1x, fpsIdx, 512, 128);
  k_gather_xyz<<<gs((long)B * 128), TPB, 0, stream>>>(l1x, fpsIdx, l2x, 512, 128, B * 128);
  {
    const float rad[2]  = {0.4f, 0.8f};
    const int   ks[2]   = {64, 128};
    const int   coff[2] = {0, 256};
    for (int br = 0; br < 2; ++br) {
      int K = ks[br], Cpad = a16(323);   // 336
      int waves = B * 128;
      k_ball_query<<<dim3((waves + 7) / 8), TPB, 0, stream>>>(
          l1x, l2x, bqIdx, 512, 128, K, rad[br] * rad[br], waves);
      long tg = (long)B * 128 * K;
      k_group<<<gs(tg), TPB, 0, stream>>>(l1x, l1p, l2x, bqIdx, gp, 512, 128, K, 320,
                                          Cpad, tg);
      int R = B * 128 * K;
      const float* res = run_mlp(gp, R, sa2m[br], 3);
      int C = sa2m[br][2].C, Cp = sa2m[br][2].Cp;
      long tp = (long)B * 128 * C;
      k_maxpool<<<gs(tp), TPB, 0, stream>>>(res, l2p, K, C, Cp, 512, coff[br], tp);
    }
  }

  // ---------- SA3 (group-all over 128 points; cin 515 -> pad 528) ----------
  k_copy_ch<<<gs((long)B * 128 * 3), TPB, 0, stream>>>(l2x, gp, 128, 128, 3, 528, 0,
                                                       (long)B * 128 * 3);
  k_copy_ch<<<gs((long)B * 128 * 512), TPB, 0, stream>>>(l2p, gp, 128, 128, 512, 528, 3,
                                                         (long)B * 128 * 512);
  k_zero_cols<<<gs((long)B * 128 * 13), TPB, 0, stream>>>(gp, 528, 515, (long)B * 128);
  {
    const float* res = run_mlp(gp, B * 128, sa3m, 3);
    long tp = (long)B * 1024;
    k_maxpool<<<gs(tp), TPB, 0, stream>>>(res, l3p, 128, 1024, 1024, 1024, 0, tp);
  }

  // ---------- FP3: concat(l2p, broadcast l3p) -> [B,128,1536] ----------
  k_copy_ch<<<gs((long)B * 128 * 512), TPB, 0, stream>>>(l2p, gp, 128, 128, 512, 1536, 0,
                                                         (long)B * 128 * 512);
  k_copy_ch<<<gs((long)B * 128 * 1024), TPB, 0, stream>>>(l3p, gp, 128, 1, 1024, 1536, 512,
                                                          (long)B * 128 * 1024);
  const float* fp3res = run_mlp(gp, B * 128, fp3m, 2);   // -> actB, [B,128,256]

  // ---------- FP2: concat(l1p, 3NN(fp3res)) -> [B,512,576] ----------
  k_copy_ch<<<gs((long)B * 512 * 320), TPB, 0, stream>>>(l1p, gp, 512, 512, 320, 576, 0,
                                                         (long)B * 512 * 320);
  k_three_nn_interp<<<gs((long)B * 512), TPB, 0, stream>>>(l1x, l2x, fp3res, gp,
                                                           512, 128, 256, fp3m[1].Cp,
                                                           576, 320, (long)B * 512);
  const float* fp2res = run_mlp(gp, B * 512, fp2m, 2);   // -> actB, [B,512,128]

  // ---------- FP1: concat(xyz, 3NN(fp2res)) -> [B,4096, 131->144] ----------
  k_copy_ch<<<gs((long)B * N * 3), TPB, 0, stream>>>(xyz_pt, gp, N, N, 3, 144, 0,
                                                     (long)B * N * 3);
  k_three_nn_interp<<<gs((long)B * N), TPB, 0, stream>>>(xyz_pt, l1x, fp2res, gp,
                                                         N, 512, 128, fp2m[1].Cp,
                                                         144, 3, (long)B * N);
  k_zero_cols<<<gs((long)B * N * 13), TPB, 0, stream>>>(gp, 144, 131, (long)B * N);
  const float* fp1res = run_mlp(gp, B * N, fp1m, 2);     // -> actB, [B,4096,128]

  // ---------- heads ----------
  run_layer(fp1res, B * N, head1, actA, true);           // bn_relu -> actA
  run_layer(actA, B * N, head2, (float*)d_out, false);   // final linear -> d_out
}